// GATLayer_29815662969289
// MI455X (gfx1250) — compile-verified
//
#include <hip/hip_runtime.h>
#include <math.h>

typedef __attribute__((ext_vector_type(2))) float v2f;
typedef __attribute__((ext_vector_type(8))) float v8f;

static constexpr int N_NODES = 50000;
static constexpr int NB      = 2;      // batch
static constexpr int DIM     = 64;     // D
static constexpr int NH      = 4;      // heads
static constexpr int DH      = 16;     // per-head dim
static constexpr int NO      = 64;     // out dim
static constexpr float SLOPE = 0.1f;

__device__ __forceinline__ float atomicMaxFloat(float* addr, float val) {
  // int/uint ordering trick; nmax initialized to -inf
  if (val >= 0.0f)
    return __int_as_float(atomicMax((int*)addr, __float_as_int(val)));
  else
    return __uint_as_float(atomicMin((unsigned int*)addr, __float_as_uint(val)));
}

// ---------------- init: rst = 0, denom = 0, nmax = -inf ----------------
__global__ void k_init(float* __restrict__ rst, float* __restrict__ denom,
                       float* __restrict__ nmax, int nRst, int nN8) {
  int t = blockIdx.x * blockDim.x + threadIdx.x;
  if (t < nRst) rst[t] = 0.0f;
  if (t < nN8) { denom[t] = 0.0f; nmax[t] = -__builtin_inff(); }
}

// ---------------- kernel 1: feat = x @ W_fc^T  (fp32 WMMA 16x16x4) -----
// x rows R = b*N + n (M dim), K = D = 64, cols = D = 64 (4 tiles of 16).
// feat stored as [n][b][64]  (flat = n*128 + b*64 + hd)
__global__ __launch_bounds__(256) void k_fc_wmma(const float* __restrict__ x,
                                                 const float* __restrict__ Wfc,
                                                 float* __restrict__ feat,
                                                 int nTiles) {
  const int wave = (int)((blockIdx.x * blockDim.x + threadIdx.x) >> 5);
  const int lane = threadIdx.x & 31;
  if (wave >= nTiles) return;                 // wave-uniform: EXEC stays all-ones
  const int r    = lane & 15;
  const int half = lane >> 4;
  const long long rowbase = (long long)wave * 16;
  const float* xrow = x + (rowbase + r) * DIM;

  v8f acc[4];
  #pragma unroll
  for (int ct = 0; ct < 4; ++ct)
    #pragma unroll
    for (int v = 0; v < 8; ++v) acc[ct][v] = 0.0f;

  #pragma unroll
  for (int k0 = 0; k0 < DIM; k0 += 4) {
    const int k = k0 + 2 * half;              // even -> 8B aligned float2
    v2f a = *(const v2f*)(xrow + k);          // A 16x4: lane r = row, half = K pair
    #pragma unroll
    for (int ct = 0; ct < 4; ++ct) {
      // B[k][j] = Wfc[j][k]  (out = X @ Wfc^T)
      v2f bf = *(const v2f*)(Wfc + (long long)(ct * 16 + r) * DIM + k);
      acc[ct] = __builtin_amdgcn_wmma_f32_16x16x4_f32(
          false, a, false, bf, (short)0, acc[ct], false, false);
    }
  }

  // C/D layout: VGPR v -> row (v + 8*half), col = r
  #pragma unroll
  for (int v = 0; v < 8; ++v) {
    long long row = rowbase + v + 8 * half;   // row = b*N + n
    int b = (int)(row / N_NODES);
    int n = (int)(row - (long long)b * N_NODES);
    float* orow = feat + ((long long)n * NB + b) * DIM + r;
    #pragma unroll
    for (int ct = 0; ct < 4; ++ct) orow[ct * 16] = acc[ct][v];
  }
}

// ---------------- kernel 2: el/er = feat . attn ------------------------
__global__ void k_attn(const float* __restrict__ feat, const float* __restrict__ al,
                       const float* __restrict__ ar, float* __restrict__ el,
                       float* __restrict__ er, int total) {
  int t = blockIdx.x * blockDim.x + threadIdx.x;   // t = n*8 + b*4 + h
  if (t >= total) return;
  int h = t & 3;
  int b = (t >> 2) & 1;
  long long n = t >> 3;
  const float* f = feat + n * 128 + b * 64 + h * 16;
  float sl = 0.f, sr = 0.f;
  #pragma unroll
  for (int j = 0; j < 16; ++j) {
    float fv = f[j];
    sl += fv * al[h * 16 + j];
    sr += fv * ar[h * 16 + j];
  }
  el[t] = sl;
  er[t] = sr;
}

// ---------------- kernel 3: edge scores + segment max ------------------
__global__ void k_edge_score(const float* __restrict__ el, const float* __restrict__ er,
                             const float* __restrict__ w, const int* __restrict__ src,
                             const int* __restrict__ dst, float* __restrict__ esc,
                             float* __restrict__ nmax, long long total) {
  long long t = (long long)blockIdx.x * blockDim.x + threadIdx.x;
  if (t >= total) return;
  long long e = t >> 3;
  int bh = (int)(t & 7);
  int s = src[e], dn = dst[e];
  float v = el[(long long)s * 8 + bh] + er[(long long)dn * 8 + bh];
  v = (v > 0.f) ? v : SLOPE * v;     // LeakyReLU
  v *= w[e];
  esc[t] = v;
  atomicMaxFloat(nmax + (long long)dn * 8 + bh, v);
}

// ---------------- kernel 4: exp + segment sum --------------------------
__global__ void k_edge_exp(float* __restrict__ esc, const float* __restrict__ nmax,
                           float* __restrict__ denom, const int* __restrict__ dst,
                           long long total) {
  long long t = (long long)blockIdx.x * blockDim.x + threadIdx.x;
  if (t >= total) return;
  long long e = t >> 3;
  int bh = (int)(t & 7);
  int dn = dst[e];
  float ex = expf(esc[t] - nmax[(long long)dn * 8 + bh]);
  esc[t] = ex;
  atomicAdd(denom + (long long)dn * 8 + bh, ex);
}

// ---------------- kernel 5: weighted aggregation -----------------------
// 128 threads/edge: k = b*64 + h*16 + dd; bh = k>>4
__global__ void k_agg(const float* __restrict__ feat, const float* __restrict__ esc,
                      const float* __restrict__ denom, const int* __restrict__ src,
                      const int* __restrict__ dst, float* __restrict__ rst,
                      long long total) {
  long long t = (long long)blockIdx.x * blockDim.x + threadIdx.x;
  if (t >= total) return;
  long long e = t >> 7;
  int k = (int)(t & 127);
  int bh = k >> 4;
  int s = src[e], dn = dst[e];
  float a = esc[e * 8 + bh] / denom[(long long)dn * 8 + bh];
  atomicAdd(rst + (long long)dn * 128 + k, feat[(long long)s * 128 + k] * a);
}

// ---------------- kernel 6: out = rst @ W_out^T + b  (fp32 WMMA) -------
// rows = n*8 + b*4 + h  (400000), K = 16, cols = 64 (4 tiles)
// output layout [B][N][H][O]
__global__ __launch_bounds__(256) void k_out_wmma(const float* __restrict__ rst,
                                                  const float* __restrict__ Wout,
                                                  const float* __restrict__ bout,
                                                  float* __restrict__ out,
                                                  int nTiles) {
  const int wave = (int)((blockIdx.x * blockDim.x + threadIdx.x) >> 5);
  const int lane = threadIdx.x & 31;
  if (wave >= nTiles) return;
  const int r    = lane & 15;
  const int half = lane >> 4;
  const long long rowbase = (long long)wave * 16;
  const float* arow = rst + (rowbase + r) * DH;

  v8f acc[4];
  #pragma unroll
  for (int ct = 0; ct < 4; ++ct) {
    float bias = bout[ct * 16 + r];           // bias depends only on col = r
    #pragma unroll
    for (int v = 0; v < 8; ++v) acc[ct][v] = bias;
  }

  #pragma unroll
  for (int k0 = 0; k0 < DH; k0 += 4) {
    const int k = k0 + 2 * half;
    v2f a = *(const v2f*)(arow + k);
    #pragma unroll
    for (int ct = 0; ct < 4; ++ct) {
      // B[k][j] = Wout[j][k]
      v2f bf = *(const v2f*)(Wout + (long long)(ct * 16 + r) * DH + k);
      acc[ct] = __builtin_amdgcn_wmma_f32_16x16x4_f32(
          false, a, false, bf, (short)0, acc[ct], false, false);
    }
  }

  #pragma unroll
  for (int v = 0; v < 8; ++v) {
    long long row = rowbase + v + 8 * half;   // row = n*8 + b*4 + h
    long long n = row >> 3;
    int b = (int)((row >> 2) & 1);
    int h = (int)(row & 3);
    float* orow = out + (((long long)b * N_NODES + n) * NH + h) * NO + r;
    #pragma unroll
    for (int ct = 0; ct < 4; ++ct) orow[ct * 16] = acc[ct][v];
  }
}

extern "C" void kernel_launch(void* const* d_in, const int* in_sizes, int n_in,
                              void* d_out, int out_size, void* d_ws, size_t ws_size,
                              hipStream_t stream) {
  // inputs: vt, x, w, src, dst, W_fc, attn_l, attn_r, W_out, b_out
  const float* x    = (const float*)d_in[1];
  const float* w    = (const float*)d_in[2];
  const int*   src  = (const int*)d_in[3];
  const int*   dst  = (const int*)d_in[4];
  const float* Wfc  = (const float*)d_in[5];
  const float* al   = (const float*)d_in[6];
  const float* ar   = (const float*)d_in[7];
  const float* Wout = (const float*)d_in[8];
  const float* bout = (const float*)d_in[9];
  const int E = in_sizes[2];

  // workspace layout (floats)
  float* ws    = (float*)d_ws;
  float* feat  = ws;                                       // N*B*64   = 6.4M
  float* el    = feat  + (size_t)N_NODES * NB * DIM;
  float* er    = el    + (size_t)N_NODES * NB * NH;        // 400k each
  float* nmax  = er    + (size_t)N_NODES * NB * NH;
  float* denom = nmax  + (size_t)N_NODES * NB * NH;
  float* rst   = denom + (size_t)N_NODES * NB * NH;        // N*128    = 6.4M
  float* esc   = rst   + (size_t)N_NODES * NB * NH * DH;   // E*8      = 6.4M

  const int TPB = 256;
  const int nRst = N_NODES * NB * NH * DH;   // 6,400,000
  const int nN8  = N_NODES * NB * NH;        //   400,000

  k_init<<<(nRst + TPB - 1) / TPB, TPB, 0, stream>>>(rst, denom, nmax, nRst, nN8);

  const int tiles1 = (N_NODES * NB + 15) / 16;       // 6250
  k_fc_wmma<<<(tiles1 * 32 + TPB - 1) / TPB, TPB, 0, stream>>>(x, Wfc, feat, tiles1);

  k_attn<<<(nN8 + TPB - 1) / TPB, TPB, 0, stream>>>(feat, al, ar, el, er, nN8);

  const long long eTot = (long long)E * 8;
  k_edge_score<<<(unsigned)((eTot + TPB - 1) / TPB), TPB, 0, stream>>>(
      el, er, w, src, dst, esc, nmax, eTot);

  k_edge_exp<<<(unsigned)((eTot + TPB - 1) / TPB), TPB, 0, stream>>>(
      esc, nmax, denom, dst, eTot);

  const long long aTot = (long long)E * 128;
  k_agg<<<(unsigned)((aTot + TPB - 1) / TPB), TPB, 0, stream>>>(
      feat, esc, denom, src, dst, rst, aTot);

  const int tiles2 = (N_NODES * NB * NH + 15) / 16;  // 25000
  k_out_wmma<<<(tiles2 * 32 + TPB - 1) / TPB, TPB, 0, stream>>>(
      rst, Wout, bout, (float*)d_out, tiles2);
}